// GNN_5720896438794
// MI455X (gfx1250) — compile-verified
//
#include <hip/hip_runtime.h>

typedef __attribute__((ext_vector_type(2))) float v2f;
typedef __attribute__((ext_vector_type(8))) float v8f;

#define BM 128
#define BN 128
#define BK 32
#define LDSS 36      // padded LDS row stride (floats) -> conflict-free fragment reads

#define Dc   1024    // feature dim / output cols
#define Nobj 256     // objects per batch
#define Rrel 2048    // relations per batch (power of two: g>>11 = batch)

// ---------------------------------------------------------------------------
// Issue the async HBM->LDS copies for one K-slab (8 x b128 per thread => 8
// ASYNCcnt increments per wave). No VGPR staging, no ds_store traffic.
// ---------------------------------------------------------------------------
__device__ __forceinline__ void async_load_slab(
    const float* __restrict__ obj, const float* __restrict__ xres,
    const int* __restrict__ edges, const float* __restrict__ W,
    float* Abuf, float* Wbuf,
    int block_row, int block_col, int K, int k0, int mode, int tid)
{
    const int seg = k0 / Dc;          // wave-uniform segment of the concat
    const int gk0 = k0 - seg * Dc;

    // A tile: BM x BK
    #pragma unroll
    for (int j = 0; j < 4; ++j) {
        int it  = tid + 256 * j;          // 0..1023
        int row = it >> 3;                // 0..127
        int c4  = (it & 7) << 2;          // 0,4,...,28
        int g   = block_row + row;
        const float* srcrow;
        if (mode == 0) {
            srcrow = ((seg == 0) ? obj : xres) + (size_t)g * Dc;
        } else if (seg == 1) {
            srcrow = xres + (size_t)g * Dc;
        } else {
            int b = g >> 11;                                  // g / Rrel
            int e = edges[g * 2 + ((seg == 0) ? 0 : 1)];      // subj / objidx
            srcrow = obj + (size_t)(b * Nobj + e) * Dc;
        }
        unsigned long long ga = (unsigned long long)(uintptr_t)(srcrow + gk0 + c4);
        unsigned           la = (unsigned)(uintptr_t)&Abuf[row * LDSS + c4];
        asm volatile("global_load_async_to_lds_b128 %0, %1, off"
                     :: "v"(la), "v"(ga) : "memory");
    }
    // W tile: BN rows of W x BK (same layout as A)
    #pragma unroll
    for (int j = 0; j < 4; ++j) {
        int it  = tid + 256 * j;
        int row = it >> 3;                // W row = output col 0..127
        int c4  = (it & 7) << 2;
        unsigned long long ga =
            (unsigned long long)(uintptr_t)(W + (size_t)(block_col + row) * K + k0 + c4);
        unsigned la = (unsigned)(uintptr_t)&Wbuf[row * LDSS + c4];
        asm volatile("global_load_async_to_lds_b128 %0, %1, off"
                     :: "v"(la), "v"(ga) : "memory");
    }
}

// ---------------------------------------------------------------------------
// Fused GEMM + gather + bias + ReLU + residual (+ mask) kernel.
// mode 0: X = [obj | xres(attr)]                   (M=8192,  K=2048)
// mode 1: X = [obj[subj] | xres(rela) | obj[oidx]] (M=65536, K=3072)
// out[m][n] = relu(sum_k X[m][k]*W[n][k] + bias[n]) + xres[m][n]  (*mask[m])
// Double-buffered LDS fed by GLOBAL_LOAD_ASYNC_TO_LDS_B128 (ASYNCcnt).
// ---------------------------------------------------------------------------
__global__ __launch_bounds__(256)
void gnn_gemm_kernel(const float* __restrict__ obj,
                     const float* __restrict__ xres,   // attr (mode0) / rela (mode1); also residual
                     const int*   __restrict__ edges,  // [M,2] (mode1)
                     const float* __restrict__ mask,   // [M]   (mode1)
                     const float* __restrict__ W,      // [Dc, K] row-major
                     const float* __restrict__ bias,   // [Dc]
                     float*       __restrict__ out,    // [M, Dc]
                     int M, int K, int mode)
{
    __shared__ __align__(16) float As[2][BM * LDSS];   // 2 x 18 KB
    __shared__ __align__(16) float Ws[2][BN * LDSS];   // 2 x 18 KB

    const int tid  = threadIdx.x;
    const int lane = tid & 31;
    const int wave = tid >> 5;        // 0..7
    const int wm   = wave >> 2;       // 0..1 : 64-row strip
    const int wn   = wave & 3;        // 0..3 : 32-col strip

    const int block_row = blockIdx.y * BM;
    const int block_col = blockIdx.x * BN;

    v8f acc[4][2] = {};               // 16x16 f32 C tiles

    // Prologue: slab 0 -> buffer 0
    async_load_slab(obj, xres, edges, W, &As[0][0], &Ws[0][0],
                    block_row, block_col, K, 0, mode, tid);

    int buf = 0;
    for (int k0 = 0; k0 < K; k0 += BK) {
        if (k0 + BK < K) {
            // issue next slab into the other buffer, then wait for current slab
            async_load_slab(obj, xres, edges, W, &As[buf ^ 1][0], &Ws[buf ^ 1][0],
                            block_row, block_col, K, k0 + BK, mode, tid);
            asm volatile("s_wait_asynccnt 0x8" ::: "memory");
        } else {
            asm volatile("s_wait_asynccnt 0x0" ::: "memory");
        }
        __syncthreads();              // all waves' asyncs for this slab landed

        const float* Acur = &As[buf][0];
        const float* Wcur = &Ws[buf][0];
        const int arow   = wm * 64 + (lane & 15);
        const int brow   = wn * 32 + (lane & 15);
        const int kphase = (lane >> 4) << 1;      // lanes 16-31 hold K+2,K+3
        #pragma unroll
        for (int kk = 0; kk < BK; kk += 4) {
            v2f afrag[4], bfrag[2];
            #pragma unroll
            for (int mi = 0; mi < 4; ++mi)
                afrag[mi] = *(const v2f*)&Acur[(arow + mi * 16) * LDSS + kk + kphase];
            #pragma unroll
            for (int ni = 0; ni < 2; ++ni)
                bfrag[ni] = *(const v2f*)&Wcur[(brow + ni * 16) * LDSS + kk + kphase];
            #pragma unroll
            for (int mi = 0; mi < 4; ++mi)
                #pragma unroll
                for (int ni = 0; ni < 2; ++ni)
                    acc[mi][ni] = __builtin_amdgcn_wmma_f32_16x16x4_f32(
                        false, afrag[mi], false, bfrag[ni],
                        (short)0, acc[mi][ni], false, false);
        }
        __syncthreads();              // buffer free before next iter overwrites it
        buf ^= 1;
    }

    // ---- epilogue: bias + ReLU + residual (+ mask), coalesced stores ----
    const int mhalf = (lane >> 4) << 3;           // lanes 16-31 hold M+8
    #pragma unroll
    for (int ni = 0; ni < 2; ++ni) {
        int n = block_col + wn * 32 + ni * 16 + (lane & 15);
        float bv = bias[n];
        #pragma unroll
        for (int mi = 0; mi < 4; ++mi) {
            int mbase = block_row + wm * 64 + mi * 16 + mhalf;
            #pragma unroll
            for (int v = 0; v < 8; ++v) {
                int m = mbase + v;
                float r = acc[mi][ni][v] + bv;
                r = fmaxf(r, 0.0f) + xres[(size_t)m * Dc + n];
                if (mode != 0) r *= mask[m];
                out[(size_t)m * Dc + n] = r;
            }
        }
    }
}

// Pass-through copy of obj_vecs (output 0), 128-bit vectorized.
__global__ __launch_bounds__(256)
void copy_obj_kernel(const float4* __restrict__ src, float4* __restrict__ dst, int n4)
{
    int i = blockIdx.x * blockDim.x + threadIdx.x;
    if (i < n4) dst[i] = src[i];
}

extern "C" void kernel_launch(void* const* d_in, const int* in_sizes, int n_in,
                              void* d_out, int out_size, void* d_ws, size_t ws_size,
                              hipStream_t stream) {
    const float* obj    = (const float*)d_in[0];   // [B,N,D]
    const float* attr   = (const float*)d_in[1];   // [B,N,D]
    const float* rela   = (const float*)d_in[2];   // [B,R,D]
    const int*   edges  = (const int*)  d_in[3];   // [B,R,2]
    const float* masks  = (const float*)d_in[4];   // [B,R,1]
    const float* W_attr = (const float*)d_in[5];   // [D, 2D]
    const float* b_attr = (const float*)d_in[6];   // [D]
    const float* W_rela = (const float*)d_in[7];   // [D, 3D]
    const float* b_rela = (const float*)d_in[8];   // [D]

    float* out = (float*)d_out;
    const int B = 32, N = Nobj, R = Rrel, D = Dc;
    const int objElems = B * N * D;                // 8,388,608
    float* out_obj  = out;
    float* out_attr = out + objElems;
    float* out_rela = out + 2 * (size_t)objElems;

    // Output 0: obj_vecs pass-through
    {
        int n4 = objElems / 4;
        copy_obj_kernel<<<(n4 + 255) / 256, 256, 0, stream>>>(
            (const float4*)obj, (float4*)out_obj, n4);
    }

    // Output 1: attr path  (M=8192, K=2048)
    {
        dim3 grid(D / BN, (B * N) / BM);
        gnn_gemm_kernel<<<grid, 256, 0, stream>>>(
            obj, attr, nullptr, nullptr, W_attr, b_attr,
            out_attr, B * N, 2 * D, /*mode=*/0);
    }

    // Output 2: rela path  (M=65536, K=3072, gathered A)
    {
        dim3 grid(D / BN, (B * R) / BM);
        gnn_gemm_kernel<<<grid, 256, 0, stream>>>(
            obj, rela, edges, masks, W_rela, b_rela,
            out_rela, B * R, 3 * D, /*mode=*/1);
    }
}